// Memristor_78030965833729
// MI455X (gfx1250) — compile-verified
//
#include <hip/hip_runtime.h>
#include <cstdint>

// Memristor recurrence on MI455X (gfx1250).
// [B=16, T=1024, C=1024] f32. One thread per (b,c) column; serial over T.
// 512 single-wave (32-thread) workgroups -> one wave per SIMD across the chip.
// Input streamed through LDS with CDNA5 async loads
// (global_load_async_to_lds_b128), 4 buffers, prefetch depth 3.

#define BLOCK 32              // one wave32 per workgroup (barriers become NOPs)
#define CHUNK 16              // timesteps per buffer (16 x 32 x 4B = 2 KB)
#define NBUF  4               // prefetch depth 3
static constexpr int Bdim = 16;
static constexpr int Tdim = 1024;
static constexpr int Cdim = 1024;

// vector-of-4-int type matching the builtin's V4i parameters
typedef __attribute__((__vector_size__(4 * sizeof(int)))) int v4i;
typedef __attribute__((address_space(1))) v4i* gv4i_ptr;   // global
typedef __attribute__((address_space(3))) v4i* lv4i_ptr;   // LDS

// ---- CDNA5 async-to-LDS helpers (builtin confirmed present on this toolchain) ----
__device__ __forceinline__ void async_load_b128(const void* g, void* l) {
#if defined(__has_builtin) && __has_builtin(__builtin_amdgcn_global_load_async_to_lds_b128)
  __builtin_amdgcn_global_load_async_to_lds_b128(
      (gv4i_ptr)(uintptr_t)g,
      (lv4i_ptr)(uint32_t)(uintptr_t)l,   // low 32 bits of flat addr = LDS byte offset
      /*offset=*/0, /*cpol=*/0);
#else
  unsigned           lds = (unsigned)(uintptr_t)l;
  unsigned long long ga  = (unsigned long long)(uintptr_t)g;
  asm volatile("global_load_async_to_lds_b128 %0, %1, off"
               :: "v"(lds), "v"(ga) : "memory");
#endif
}

#if defined(__has_builtin) && __has_builtin(__builtin_amdgcn_s_wait_asynccnt)
#define WAIT_ASYNC(n) do { __builtin_amdgcn_s_wait_asynccnt(n); \
                           asm volatile("" ::: "memory"); } while (0)
#else
#define WAIT_ASYNC(n) asm volatile("s_wait_asynccnt %0" :: "n"(n) : "memory")
#endif

// ---- one memristor timestep ----
// Invariants used to drop dead clamps: res >= 0, fil >= 0, res + fil <= 1.
__device__ __forceinline__ float mstep(float vin, float& res, float& fil, float& S) {
  const float LOG2E = 1.44269504088896340736f;
  // v = vin / Rm * Rm == vin exactly in real arithmetic (R_Compliance = 0)
  float tot   = fminf(res + fil, 1.0f);
  bool  smask = tot > 0.999999f;
  bool  cmask = S   < 1.000001f;
  float vpos  = fmaxf(vin, 0.0f);

  // dS = (THETA*vpos*exp(1-S) - GAMMA3*exp(S-1)) * DT ; exp(S-1) = 2^(-u)
  float u  = (1.0f - S) * LOG2E;
  float e1 = __builtin_amdgcn_exp2f(u);
  float e2 = __builtin_amdgcn_exp2f(-u);
  float dS = (vpos * e1 - 5.0f * e2) * 0.001f;           // THETA=1, GAMMA3=5, DT=1e-3
  S = fmaxf(S + (smask ? dS : 0.0f), 1.0f);

  // diffusion: fil -= clip(0.6*fil,0,fil)*cmask == fil *= 0.4 ; res *= 0.998
  fil *= cmask ? 0.4f   : 1.0f;
  res *= cmask ? 0.998f : 1.0f;

  // drift: g = clip(ALPHA+(1-tot), ALPHA, ALPHA+1) == 1.01 - tot  (tot in [0,1])
  float g     = 1.01f - tot;
  float drift = fminf(vpos * 0.22f * __builtin_amdgcn_rcpf(g),   // MU*DT = 0.22
                      1.0f - fil - res);
  fil += cmask ? drift : 0.0f;

  // transform = clip(BETA*DT*fil, 0, min(1-res, fil)) == min(0.01*fil, 1-res)
  float tr = fminf(0.01f * fil, 1.0f - res);
  float tm = cmask ? tr : 0.0f;
  fil -= tm;
  res += tm;

  float tot2 = fminf(res + fil, 1.0f);
  // Rm = RON*tot2/S^2 + ROFF*(exp((1-tot2)/LAM)-1)/(exp(1/LAM)-1)
  float ex = __builtin_amdgcn_exp2f((1.0f - tot2) * (5.0f * LOG2E));
  float Rm = 1.0e7f * tot2 * __builtin_amdgcn_rcpf(S * S)
           + (float)(1.0e12 / 147.41315910257660342) * (ex - 1.0f);
  return vin * __builtin_amdgcn_rcpf(Rm);               // current = vin / Rm
}

__global__ __launch_bounds__(BLOCK)
void memristor_scan_kernel(const float* __restrict__ Vin, float* __restrict__ Iout) {
  __shared__ float lbuf[NBUF][CHUNK][BLOCK];            // 4 x 2 KB ring

  const int tid = threadIdx.x;                          // 0..31 (one wave)
  const int b   = blockIdx.x >> 5;                      // 32 blocks per batch row
  const int cb  = (blockIdx.x & 31) * BLOCK;
  const float* __restrict__ src = Vin  + (size_t)b * Tdim * Cdim + cb;
  float*       __restrict__ dst = Iout + (size_t)b * Tdim * Cdim + cb;

  // one CHUNK (16 rows x 128 B) = 4 async b128 loads per lane
  auto issue_chunk = [&](int t0, int buf) {
#pragma unroll
    for (int p = 0; p < 4; ++p) {
      int j   = p * BLOCK + tid;                        // 0..127
      int r   = j >> 3;                                 // row 0..15
      int seg = (j & 7) << 2;                           // float index 0..28
      async_load_b128(src + (size_t)(t0 + r) * Cdim + seg, &lbuf[buf][r][seg]);
    }
  };

  float res = 0.0f, fil = 0.0f, S = 0.0f;               // Rm drops out of the state

  auto compute_chunk = [&](int k, int buf) {
#pragma unroll
    for (int r = 0; r < CHUNK; ++r) {
      float cur = mstep(lbuf[buf][r][tid], res, fil, S);
      dst[(size_t)(k * CHUNK + r) * Cdim + tid] = cur;  // coalesced global_store_b32
    }
  };

  const int nchunk = Tdim / CHUNK;                      // 64
  issue_chunk(0 * CHUNK, 0);                            // prime 3 chunks
  issue_chunk(1 * CHUNK, 1);
  issue_chunk(2 * CHUNK, 2);

  for (int k = 0; k < nchunk - 3; ++k) {
    issue_chunk((k + 3) * CHUNK, (k + 3) & (NBUF - 1));
    WAIT_ASYNC(12);                                     // chunk k landed (3 ahead in flight)
    compute_chunk(k, k & (NBUF - 1));
  }
  WAIT_ASYNC(8);  compute_chunk(nchunk - 3, (nchunk - 3) & (NBUF - 1));
  WAIT_ASYNC(4);  compute_chunk(nchunk - 2, (nchunk - 2) & (NBUF - 1));
  WAIT_ASYNC(0);  compute_chunk(nchunk - 1, (nchunk - 1) & (NBUF - 1));
}

extern "C" void kernel_launch(void* const* d_in, const int* in_sizes, int n_in,
                              void* d_out, int out_size, void* d_ws, size_t ws_size,
                              hipStream_t stream) {
  (void)in_sizes; (void)n_in; (void)out_size; (void)d_ws; (void)ws_size;
  const float* Vin = (const float*)d_in[0];
  float*       out = (float*)d_out;
  dim3 grid(Bdim * Cdim / BLOCK);                       // 512 blocks x 32 threads
  memristor_scan_kernel<<<grid, BLOCK, 0, stream>>>(Vin, out);
}